// RFNN_Encoder_64991445123988
// MI455X (gfx1250) — compile-verified
//
#include <hip/hip_runtime.h>

// RFNN scan: m_{t+1}[r] = normalize_r( prod_f( exp(-((x_t[f]-c[r,f])/s[r,f])^2/2) + m_t[r] ) )
// Strictly serial over t=0..4095 -> single persistent workgroup, LDS-only cross-rule exchange.
// Per-step per-thread hot path: 4x global_load_b128, 16x (2 FMA + v_exp_f32 + add + mul),
// 4x width-16 xor-shuffle product, 1 split barrier, 5x width-32 xor-shuffle sum, rcp+NR normalize.

#define SEQ_LEN 4096
#define FEATS   256
#define RULES   64
#define TPR     16                 // threads per rule
#define FPT     16                 // features per thread
#define TPB     (RULES * TPR)      // 1024 threads = 32 wave32 waves on one WGP

__launch_bounds__(TPB, 1)
__global__ void rfnn_scan_kernel(const float* __restrict__ x,
                                 const float* __restrict__ center,
                                 const float* __restrict__ sigma,
                                 float* __restrict__ out) {
    // double-buffered rule values: one s_barrier per timestep
    __shared__ float lds_rule[2][RULES];

    const int tid  = threadIdx.x;
    const int r    = tid >> 4;     // rule id 0..63
    const int j    = tid & 15;     // feature-group id 0..15 (within-wave)
    const int lane = tid & 31;     // wave32 lane

    // One-time: pull this thread's 16 gaussian params into VGPRs as a quadratic form.
    // exp(-d^2/(2 s^2)) == exp2(n*(x-c)^2) with n = -log2(e)/(2 s^2)
    //                   == exp2(A*x^2 + B*x + C),  A = n, B = -2*n*c, C = n*c^2
    float A[FPT], B[FPT], C[FPT];
    {
        const float* cb = center + r * FEATS + j * FPT;
        const float* sb = sigma  + r * FEATS + j * FPT;
        #pragma unroll
        for (int k = 0; k < FPT; ++k) {
            const float c = cb[k];
            const float s = sb[k];
            const float n = -0.72134752f / (s * s);   // one-time IEEE div, cost irrelevant
            A[k] = n;
            B[k] = -2.0f * n * c;
            C[k] = n * c * c;
        }
    }

    float m = 0.0f;  // carried memory for rule r (replicated across the 16 lanes of the group)

    for (int t = 0; t < SEQ_LEN; ++t) {
        // This thread's 16 contiguous features of row t: 4x global_load_b128 (row is 1KB, lives in WGP$)
        const float4* xv4 = reinterpret_cast<const float4*>(x + (size_t)t * FEATS) + j * 4;
        if (t + 1 < SEQ_LEN) {
            // gfx1250 global_prefetch_b8 on next row while we compute this one
            __builtin_prefetch(x + (size_t)(t + 1) * FEATS + j * FPT, 0, 3);
        }

        float p = 1.0f;
        #pragma unroll
        for (int k = 0; k < 4; ++k) {
            const float4 xv = xv4[k];
            const float a0 = fmaf(fmaf(A[4*k+0], xv.x, B[4*k+0]), xv.x, C[4*k+0]);
            const float a1 = fmaf(fmaf(A[4*k+1], xv.y, B[4*k+1]), xv.y, C[4*k+1]);
            const float a2 = fmaf(fmaf(A[4*k+2], xv.z, B[4*k+2]), xv.z, C[4*k+2]);
            const float a3 = fmaf(fmaf(A[4*k+3], xv.w, B[4*k+3]), xv.w, C[4*k+3]);
            p *= __builtin_amdgcn_exp2f(a0) + m;   // v_exp_f32
            p *= __builtin_amdgcn_exp2f(a1) + m;
            p *= __builtin_amdgcn_exp2f(a2) + m;
            p *= __builtin_amdgcn_exp2f(a3) + m;
        }

        // Product across the 16 lanes of this rule's group (stays inside one wave32)
        p *= __shfl_xor(p, 1, 16);
        p *= __shfl_xor(p, 2, 16);
        p *= __shfl_xor(p, 4, 16);
        p *= __shfl_xor(p, 8, 16);

        const int buf = t & 1;
        if (j == 0) lds_rule[buf][r] = p;
        __syncthreads();  // split barrier: s_barrier_signal -1 / s_barrier_wait -1

        // Every wave redundantly computes the sum over 64 rules (identical op order ->
        // bitwise-identical in all waves, so no second barrier; double buffering handles
        // the WAR hazard on lds_rule across iterations).
        float a = lds_rule[buf][lane] + lds_rule[buf][lane + 32];
        a += __shfl_xor(a, 1, 32);
        a += __shfl_xor(a, 2, 32);
        a += __shfl_xor(a, 4, 32);
        a += __shfl_xor(a, 8, 32);
        a += __shfl_xor(a, 16, 32);

        // Normalize: v_rcp_f32 + one Newton-Raphson step (~0.5 ulp) instead of the
        // full IEEE div expansion; the reference's s==0 guard kept as a select.
        float inv = __builtin_amdgcn_rcpf(a);
        inv = inv * (2.0f - a * inv);

        const float rv = lds_rule[buf][r];   // LDS broadcast within the group
        m = (a == 0.0f) ? rv : rv * inv;
    }

    if (j == 0) out[r] = m;
}

extern "C" void kernel_launch(void* const* d_in, const int* in_sizes, int n_in,
                              void* d_out, int out_size, void* d_ws, size_t ws_size,
                              hipStream_t stream) {
    (void)in_sizes; (void)n_in; (void)d_ws; (void)ws_size; (void)out_size;
    const float* x      = (const float*)d_in[0];  // (4096, 256) f32
    const float* center = (const float*)d_in[1];  // (64, 256)  f32
    const float* sigma  = (const float*)d_in[2];  // (64, 256)  f32
    float* out = (float*)d_out;                   // (64,)      f32

    rfnn_scan_kernel<<<1, TPB, 0, stream>>>(x, center, sigma, out);
}